// EmbedAtt_29695403885264
// MI455X (gfx1250) — compile-verified
//
#include <hip/hip_runtime.h>

typedef __attribute__((ext_vector_type(2))) float v2f;
typedef __attribute__((ext_vector_type(8))) float v8f;

// Problem constants (fixed by the reference)
constexpr int BROWS = 65536;
constexpr int CN    = 16;
constexpr int CC    = 16;
constexpr int DDIM  = 128;
constexpr int VCAT  = 65;
constexpr float EPS = 1e-5f;

constexpr int ROWS_PER_BLOCK = 128;   // 8 waves x 16 rows
constexpr int THREADS        = 256;   // 8 wave32s
constexpr int LWS            = DDIM + 8;  // padded LDS stride for lin_W (bank stagger)

__global__ __launch_bounds__(THREADS)
void embed_att_kernel(const float* __restrict__ x_num,
                      const float* __restrict__ means,
                      const float* __restrict__ stds,
                      const float* __restrict__ lin_W,
                      const float* __restrict__ lin_b,
                      const float* __restrict__ emb,
                      const int*   __restrict__ x_cat,
                      float* __restrict__ out) {
  __shared__ float s_lw[CN * LWS];                  // ~8.5 KB, staged lin_W (padded rows)
  __shared__ float s_bias[DDIM];                    // 512 B, sum_k lin_b[k][:]
  __shared__ float s_ivs[CN];                       // 64 B, 1/(std+eps) per feature
  __shared__ float s_h[ROWS_PER_BLOCK * DDIM];      // 64 KB, WMMA result tile

  const int tid      = threadIdx.x;
  const int lane     = tid & 31;
  const int wave     = tid >> 5;
  const int rowBlock = blockIdx.x * ROWS_PER_BLOCK;

  // ---- Phase 0: stage lin_W (padded), bias-sum, inv-std into LDS (coalesced) ----
  #pragma unroll
  for (int i = 0; i < (CN * DDIM) / THREADS; ++i) {
    const int e = i * THREADS + tid;
    const int k = e >> 7;            // /128
    const int d = e & (DDIM - 1);
    s_lw[k * LWS + d] = lin_W[e];
  }
  if (tid < DDIM) {
    float b = 0.f;
    #pragma unroll
    for (int k = 0; k < CN; ++k) b += lin_b[k * DDIM + tid];
    s_bias[tid] = b;
  }
  if (tid < CN) {
    // precise division, once per block: exact per-feature reciprocal scale
    s_ivs[tid] = 1.0f / (stds[tid] + EPS);
  }
  __syncthreads();

  // ---- Phase 1: per-wave f32 WMMA: 16 rows x (8 tiles of N=16), K=16 as 4x K=4 ----
  {
    const int m   = lane & 15;       // row within tile (A) / column (B)
    const int hi  = lane >> 4;       // 0: K pair {0,1} of each chunk ; 1: {2,3}
    const int row = rowBlock + wave * 16 + m;

    // A values: z = sigmoid((x - mean) * invstd), f32, two K-values per chunk
    float a0[4], a1[4];
    #pragma unroll
    for (int q = 0; q < 4; ++q) {
      const int ko = 4 * q + 2 * hi;
      const float2 xv = *(const float2*)(x_num + (size_t)row * CN + ko);
      const float2 mv = *(const float2*)(means + ko);
      const float i0 = s_ivs[ko];
      const float i1 = s_ivs[ko + 1];
      const float t0 = (xv.x - mv.x) * i0;
      const float t1 = (xv.y - mv.y) * i1;
      // sigmoid via fast reciprocal (v_rcp_f32, ~1 ulp)
      a0[q] = __builtin_amdgcn_rcpf(1.0f + __expf(-t0));
      a1[q] = __builtin_amdgcn_rcpf(1.0f + __expf(-t1));
    }

    const int n     = lane & 15;
    const int koff  = hi * 2;                      // B rows: lanes0-15 -> k,k+1 ; lanes16-31 -> k+2,k+3
    const int rbase = wave * 16 + (hi ? 8 : 0);    // C/D layout row base

    #pragma unroll
    for (int t = 0; t < DDIM / 16; ++t) {
      v8f c;
      #pragma unroll
      for (int j = 0; j < 8; ++j) c[j] = 0.f;

      #pragma unroll
      for (int q = 0; q < 4; ++q) {
        v2f a;
        a[0] = a0[q];
        a[1] = a1[q];
        v2f b;
        b[0] = s_lw[(4 * q + koff + 0) * LWS + t * 16 + n];
        b[1] = s_lw[(4 * q + koff + 1) * LWS + t * 16 + n];
        c = __builtin_amdgcn_wmma_f32_16x16x4_f32(
                /*neg_a=*/false, a, /*neg_b=*/false, b,
                /*c_mod=*/(short)0, c, /*reuse_a=*/false, /*reuse_b=*/false);
      }

      // Scatter C fragment into LDS tile: lane<16 -> rows rbase+0..7, lane>=16 -> +8..15
      #pragma unroll
      for (int j = 0; j < 8; ++j)
        s_h[(rbase + j) * DDIM + t * 16 + n] = c[j];
    }
  }
  __syncthreads();

  // ---- Phase 2: embedding gather + bias + h, fully coalesced output ----
  {
    const int d     = tid & (DDIM - 1);
    const int rhalf = tid >> 7;                 // 256 threads cover 2 rows/iter
    #pragma unroll 1
    for (int it = 0; it < ROWS_PER_BLOCK / 2; ++it) {
      const int lr  = it * 2 + rhalf;
      const int row = rowBlock + lr;

      // Prefetch upcoming category-index rows (global_prefetch_b8)
      __builtin_prefetch(x_cat + (size_t)(row + 4) * CC, 0, 1);

      const int4* ci = (const int4*)(x_cat + (size_t)row * CC);
      const int4 c0 = ci[0], c1 = ci[1], c2 = ci[2], c3 = ci[3];
      const int idxs[16] = {c0.x, c0.y, c0.z, c0.w,
                            c1.x, c1.y, c1.z, c1.w,
                            c2.x, c2.y, c2.z, c2.w,
                            c3.x, c3.y, c3.z, c3.w};

      float acc = s_h[lr * DDIM + d] + s_bias[d];
      #pragma unroll
      for (int c = 0; c < CC; ++c)
        acc += emb[((size_t)(c * VCAT) + idxs[c]) * DDIM + d];

      out[(size_t)row * DDIM + d] = acc;
    }
  }
}

extern "C" void kernel_launch(void* const* d_in, const int* in_sizes, int n_in,
                              void* d_out, int out_size, void* d_ws, size_t ws_size,
                              hipStream_t stream) {
  (void)in_sizes; (void)n_in; (void)out_size; (void)d_ws; (void)ws_size;
  const float* x_num = (const float*)d_in[0];
  const float* means = (const float*)d_in[1];
  const float* stds  = (const float*)d_in[2];
  const float* lin_W = (const float*)d_in[3];
  const float* lin_b = (const float*)d_in[4];
  const float* emb   = (const float*)d_in[5];
  const int*   x_cat = (const int*)  d_in[6];
  float*       out   = (float*)d_out;

  dim3 grid(BROWS / ROWS_PER_BLOCK);
  dim3 block(THREADS);
  embed_att_kernel<<<grid, block, 0, stream>>>(x_num, means, stds, lin_W, lin_b,
                                               emb, x_cat, out);
}